// StackedSIGN_57397942944432
// MI455X (gfx1250) — compile-verified
//
#include <hip/hip_runtime.h>
#include <stdint.h>

typedef __attribute__((ext_vector_type(2))) float v2f;
typedef __attribute__((ext_vector_type(8))) float v8f;
typedef __attribute__((ext_vector_type(4))) uint32_t u32x4;
typedef __attribute__((ext_vector_type(4))) int i32x4v;
typedef __attribute__((ext_vector_type(8))) int i32x8v;

#define CIN   128
#define COUT  64
#define NHOP  3              // feature sets: x, Ax, A^2x
#define PSTR  132            // padded row stride (floats) per K-pair row
#define NW    (64 * PSTR)    // dwords per packed per-hop weight block (8448)

#if defined(__AMDGCN__) && __has_builtin(__builtin_amdgcn_tensor_load_to_lds)
  #define HAVE_TDM 1
  #if __has_include(<hip/amd_detail/amd_gfx1250_TDM.h>)
    #define TDM_6ARG 1
  #else
    #define TDM_6ARG 0
  #endif
#else
  #define HAVE_TDM 0
#endif

// ---------------------------------------------------------------------------
// Zero the two hop buffers in workspace (atomics accumulate into them).
// ---------------------------------------------------------------------------
__global__ __launch_bounds__(256) void zero_ws(float* __restrict__ p, int n4) {
  int i = blockIdx.x * 256 + threadIdx.x;
  if (i < n4) ((float4*)p)[i] = make_float4(0.f, 0.f, 0.f, 0.f);
}

// ---------------------------------------------------------------------------
// Pack W1 (3 x 128 x 64, row-major [k][n]) into K-pair-interleaved, padded
// layout so each WMMA B fragment is one contiguous 8-byte LDS load:
//   Wp[hop][k/2][n*2 + (k&1)]  , row stride PSTR floats (bank skew).
// This global image is copied verbatim into LDS by the TDM.
// ---------------------------------------------------------------------------
__global__ __launch_bounds__(256) void pack_w(const float* __restrict__ W1,
                                              float* __restrict__ Wp) {
  int idx = blockIdx.x * 256 + threadIdx.x;
  if (idx >= NHOP * CIN * COUT) return;
  int hop = idx / (CIN * COUT);
  int r   = idx - hop * (CIN * COUT);
  int k   = r >> 6;
  int n   = r & 63;
  Wp[hop * NW + (k >> 1) * PSTR + (n << 1) + (k & 1)] = W1[idx];
}

// ---------------------------------------------------------------------------
// COO scatter-add SpMM: dst[row[e], :] += src[col[e], :]
// 32 threads per edge, float4 gather + 4 hardware f32 atomics (L2-resident).
// ---------------------------------------------------------------------------
__global__ __launch_bounds__(256) void spmm_scatter(
    const int64_t* __restrict__ rows, const int64_t* __restrict__ cols,
    const float* __restrict__ src, float* __restrict__ dst, int E)
{
  int tid = blockIdx.x * 256 + threadIdx.x;
  int e = tid >> 5;
  if (e >= E) return;
  int c4 = (tid & 31) << 2;
  int64_t r = rows[e];
  int64_t c = cols[e];
  const float4 v = *(const float4*)(src + c * CIN + c4);
  float* d = dst + r * CIN + c4;
  unsafeAtomicAdd(d + 0, v.x);
  unsafeAtomicAdd(d + 1, v.y);
  unsafeAtomicAdd(d + 2, v.z);
  unsafeAtomicAdd(d + 3, v.w);
}

// ---------------------------------------------------------------------------
// out[n,o] = sum_k xs[k][n,:] @ W1[k][:,o] + sum_k b1[k][o]
// One wave computes a 16(M) x 64(N) tile via V_WMMA_F32_16X16X4_F32.
// 5 waves / block, 625 blocks -> exactly 3125 M-tiles (50000 nodes).
// Per-hop packed weight block (33792 B) staged into LDS by the Tensor Data
// Mover (tensor_load_to_lds + s_wait_tensorcnt), with a synchronous fallback.
//
// Fragment layouts (ISA 7.12.2, 32-bit, wave32):
//   A 16x4 : lane L holds row M=L%16; VGPR0/1 = K = (L/16)*2 + {0,1}
//   B 4x16 : VGPR0 = K rows {0,2} split across half-waves, N = L%16;
//            VGPR1 = K rows {1,3}  -> contiguous v2f in packed layout
//   C/D    : VGPR v holds M = v (lanes 0-15) / v+8 (lanes 16-31), N = L%16
// ---------------------------------------------------------------------------
__global__ __launch_bounds__(160) void sign_gemm(
    const float* __restrict__ x0, const float* __restrict__ x1,
    const float* __restrict__ x2, const float* __restrict__ Wp,
    const float* __restrict__ b1, float* __restrict__ out, int nodes)
{
  __shared__ float Wlds[NW];   // 33792 B: one hop's packed weight block

  const int t     = threadIdx.x;
  const int wave  = t >> 5;
  const int lane  = t & 31;
  const int mtiles = nodes >> 4;
  const int mtile  = blockIdx.x * 5 + wave;
  const bool live  = (mtile < mtiles);         // wave-uniform
  const int m0     = (live ? mtile : 0) << 4;  // clamp: reads stay in-bounds
  const int row    = m0 + (lane & 15);
  const int khalf  = (lane >> 4) << 1;         // 0 or 2
  const int ncol   = lane & 31 & 15;

  v8f acc[4] = {};

  const float* feats[NHOP] = {x0, x1, x2};
#pragma unroll
  for (int hop = 0; hop < NHOP; ++hop) {
    __syncthreads();   // previous hop's LDS reads complete in all waves
#if HAVE_TDM
    if (t == 0) {
      uint64_t ga = (uint64_t)(uintptr_t)(Wp + hop * NW);
      uint32_t la = (uint32_t)(uintptr_t)(&Wlds[0]);   // LDS byte offset
      u32x4 g0;
      g0[0] = 1u;                                        // count=1, user D#
      g0[1] = la;                                        // lds_addr
      g0[2] = (uint32_t)ga;                              // global_addr[31:0]
      g0[3] = (uint32_t)((ga >> 32) & 0x01FFFFFFu) | (2u << 30); // [56:32]|type=2
      i32x8v g1;
      g1[0] = (int)(2u << 16);                           // data_size = 4B
      g1[1] = (int)((NW & 0xFFFFu) << 16);               // tensor_dim0[15:0]
      g1[2] = (int)(((uint32_t)NW >> 16) | (1u << 16));  // dim0[31:16], dim1=1
      g1[3] = (int)((NW & 0xFFFFu) << 16);               // tile_dim0 = NW
      g1[4] = 1;                                         // tile_dim1 = 1
      g1[5] = (int)NW;                                   // dim0_stride[31:0]
      g1[6] = 0;
      g1[7] = 0;
      i32x4v gz = {0, 0, 0, 0};
  #if TDM_6ARG
      i32x8v gz8 = {0, 0, 0, 0, 0, 0, 0, 0};
      __builtin_amdgcn_tensor_load_to_lds(g0, g1, gz, gz, gz8, 0);
  #else
      __builtin_amdgcn_tensor_load_to_lds(g0, g1, gz, gz, 0);
  #endif
    }
    __builtin_amdgcn_s_wait_tensorcnt(0);  // issuing wave drains TDM
    __syncthreads();                       // publish LDS to all waves
#else
    {
      const float4* wsrc = (const float4*)(Wp + hop * NW);
      for (int i = t; i < NW / 4; i += 160)
        ((float4*)Wlds)[i] = wsrc[i];
    }
    __syncthreads();
#endif

    const float* frow = feats[hop] + (size_t)row * CIN;
    for (int kl = 0; kl < CIN; kl += 4) {
      // A fragment: two consecutive K values for this lane's row (8B aligned)
      v2f a = *(const v2f*)(frow + kl + khalf);
      // B fragments: one contiguous v2f per N-tile in packed layout
      const float* wp = Wlds + ((kl >> 1) + (lane >> 4)) * PSTR + (ncol << 1);
#pragma unroll
      for (int nt = 0; nt < 4; ++nt) {
        v2f b = *(const v2f*)(wp + nt * 32);
        acc[nt] = __builtin_amdgcn_wmma_f32_16x16x4_f32(
            false, a, false, b, (short)0, acc[nt], false, false);
      }
    }
  }

  if (!live) return;
#pragma unroll
  for (int nt = 0; nt < 4; ++nt) {
    int c = nt * 16 + ncol;
    float bs = b1[c] + b1[COUT + c] + b1[2 * COUT + c];
#pragma unroll
    for (int v = 0; v < 8; ++v) {
      int r = m0 + v + ((lane >> 4) << 3);
      out[(size_t)r * COUT + c] = acc[nt][v] + bs;
    }
  }
}

// ---------------------------------------------------------------------------
// Inputs (setup_inputs order):
//  0: x  (N*128 f32)   1: W0 (dead)  2: b0 (dead)
//  3: W1 (3*128*64 f32) 4: b1 (3*64 f32)
//  5: edge_index (2*E int64)  6: batch (dead)
// Output: (N*64 f32)
// Workspace: h1 (N*128 f32) | h2 (N*128 f32) | Wp (3*8448 f32)
// ---------------------------------------------------------------------------
extern "C" void kernel_launch(void* const* d_in, const int* in_sizes, int n_in,
                              void* d_out, int out_size, void* d_ws, size_t ws_size,
                              hipStream_t stream) {
  const float*   x   = (const float*)d_in[0];
  const float*   W1  = (const float*)d_in[3];
  const float*   b1  = (const float*)d_in[4];
  const int64_t* ei  = (const int64_t*)d_in[5];
  float*         out = (float*)d_out;

  const int nodes = in_sizes[0] / CIN;   // 50000
  const int E     = in_sizes[5] / 2;     // 800000

  float* h1 = (float*)d_ws;                       // A x    (N x 128)
  float* h2 = h1 + (size_t)nodes * CIN;           // A^2 x  (N x 128)
  float* Wp = h2 + (size_t)nodes * CIN;           // packed weights

  const int64_t* rows = ei;        // edge_index[0]
  const int64_t* cols = ei + E;    // edge_index[1]

  // 1) zero hop buffers; pack weights
  int n4 = (nodes * CIN * 2) / 4;
  zero_ws<<<(n4 + 255) / 256, 256, 0, stream>>>(h1, n4);
  pack_w<<<(NHOP * CIN * COUT + 255) / 256, 256, 0, stream>>>(W1, Wp);

  // 2) hop 1: h1 = A x ; hop 2: h2 = A h1
  int sthreads = E * 32;
  int sblocks  = (sthreads + 255) / 256;
  spmm_scatter<<<sblocks, 256, 0, stream>>>(rows, cols, x,  h1, E);
  spmm_scatter<<<sblocks, 256, 0, stream>>>(rows, cols, h1, h2, E);

  // 3) fused 3-hop GEMM + bias
  int mtiles = nodes >> 4;
  int gblocks = (mtiles + 4) / 5;   // 625 for N=50000
  sign_gemm<<<gblocks, 160, 0, stream>>>(x, h1, h2, Wp, b1, out, nodes);
}